// VMDecoding_69423851372724
// MI455X (gfx1250) — compile-verified
//
#include <hip/hip_runtime.h>

// CDNA5 (gfx1250) wave32 WMMA types
typedef __attribute__((ext_vector_type(2))) float v2f;   // A/B of 16x16x4 f32 WMMA: 64 vals / 32 lanes
typedef __attribute__((ext_vector_type(8))) float v8f;   // C/D 16x16 f32: 256 vals / 32 lanes

#define RES 300
#define NCH 24
#define HW  (RES * RES)

// ---------------------------------------------------------------------------
// Repack a plane from (C,H,W) channel-major to (H,W,C) texel-major so the 24
// channels of one bilinear corner are 96 contiguous bytes -> b128 loads.
// ---------------------------------------------------------------------------
__global__ __launch_bounds__(256) void vm_repack_plane_kernel(const float* __restrict__ src,
                                                              float* __restrict__ dst, int n) {
    int idx = blockIdx.x * blockDim.x + threadIdx.x;   // over C*HW, coalesced read
    if (idx < n) {
        int c = idx / HW;
        int t = idx - c * HW;
        dst[t * NCH + c] = src[idx];
    }
}

// Repack a line from (C,L) to (L,C): channels of one tap contiguous.
__global__ __launch_bounds__(256) void vm_repack_line_kernel(const float* __restrict__ src,
                                                             float* __restrict__ dst, int n) {
    int idx = blockIdx.x * blockDim.x + threadIdx.x;   // over C*L
    if (idx < n) {
        int c = idx / RES;
        int l = idx - c * RES;
        dst[l * NCH + c] = src[idx];
    }
}

// ---------------------------------------------------------------------------
// One tri-plane term: bilinear plane sample * linear line sample, channel
// products fed into V_WMMA_F32_16X16X4_F32 with B == ones => row-sum reduce.
//
// PACKED channel partition (valid because the channel sum is order-invariant):
//   j = 0..2, each lane loads float4 of channels [8j+4h .. 8j+4h+3] per corner.
//   WMMA chunk 2j   : lanes h=0 give K0,K1 = ch 8j+0,8j+1 ; h=1 give K2,K3 = ch 8j+4,8j+5
//   WMMA chunk 2j+1 : lanes h=0 give K0,K1 = ch 8j+2,8j+3 ; h=1 give K2,K3 = ch 8j+6,8j+7
// -> 3 b128 loads per corner per lane instead of 6 b64.
// ---------------------------------------------------------------------------
template <bool PACKED>
__device__ __forceinline__ v8f vm_term(v8f acc, v2f ones,
                                       const float* __restrict__ plane,
                                       const float* __restrict__ line,
                                       float u, float v, float t, int h) {
    const float s = 0.5f * (RES - 1);
    // bilinear setup (matches reference: clamp floor, weight vs clamped index)
    float ix = fmaf(u, s, s);
    float iy = fmaf(v, s, s);
    int x0 = min(max((int)floorf(ix), 0), RES - 1);
    int y0 = min(max((int)floorf(iy), 0), RES - 1);
    int x1 = min(x0 + 1, RES - 1);
    int y1 = min(y0 + 1, RES - 1);
    float wx = ix - (float)x0;
    float wy = iy - (float)y0;
    float w00 = (1.f - wx) * (1.f - wy);
    float w01 = wx * (1.f - wy);
    float w10 = (1.f - wx) * wy;
    float w11 = wx * wy;
    int t00 = y0 * RES + x0;
    int t01 = y0 * RES + x1;
    int t10 = y1 * RES + x0;
    int t11 = y1 * RES + x1;
    // line setup
    float it = fmaf(t, s, s);
    int i0 = min(max((int)floorf(it), 0), RES - 1);
    int i1 = min(i0 + 1, RES - 1);
    float wt = it - (float)i0;

    if constexpr (PACKED) {
        const float4* b00 = (const float4*)(plane + (size_t)t00 * NCH);
        const float4* b01 = (const float4*)(plane + (size_t)t01 * NCH);
        const float4* b10 = (const float4*)(plane + (size_t)t10 * NCH);
        const float4* b11 = (const float4*)(plane + (size_t)t11 * NCH);
        const float4* lb0 = (const float4*)(line + (size_t)i0 * NCH);
        const float4* lb1 = (const float4*)(line + (size_t)i1 * NCH);
        float wt0 = 1.f - wt;
#pragma unroll
        for (int j = 0; j < 3; ++j) {
            int ci = 2 * j + h;                // float4 slot: channels 8j+4h .. 8j+4h+3
            float4 p00 = b00[ci];
            float4 p01 = b01[ci];
            float4 p10 = b10[ci];
            float4 p11 = b11[ci];
            float4 q0  = lb0[ci];
            float4 q1  = lb1[ci];
            float sx = w00 * p00.x + w01 * p01.x + w10 * p10.x + w11 * p11.x;
            float sy = w00 * p00.y + w01 * p01.y + w10 * p10.y + w11 * p11.y;
            float sz = w00 * p00.z + w01 * p01.z + w10 * p10.z + w11 * p11.z;
            float sw = w00 * p00.w + w01 * p01.w + w10 * p10.w + w11 * p11.w;
            float Lx = q0.x * wt0 + q1.x * wt;
            float Ly = q0.y * wt0 + q1.y * wt;
            float Lz = q0.z * wt0 + q1.z * wt;
            float Lw = q0.w * wt0 + q1.w * wt;
            v2f a0, a1;
            a0.x = sx * Lx;
            a0.y = sy * Ly;
            a1.x = sz * Lz;
            a1.y = sw * Lw;
            // D[m][n] = sum_k A[m][k] * 1 + C[m][n]  (row-sum accumulate)
            acc = __builtin_amdgcn_wmma_f32_16x16x4_f32(
                false, a0, false, ones, (short)0, acc, false, false);
            acc = __builtin_amdgcn_wmma_f32_16x16x4_f32(
                false, a1, false, ones, (short)0, acc, false, false);
        }
    } else {
        // Fallback: original (C,H,W) planes and (C,L) lines, scalar gathers.
        const float* b00 = plane + t00;
        const float* b01 = plane + t01;
        const float* b10 = plane + t10;
        const float* b11 = plane + t11;
        const float* l0 = line + i0;
        const float* l1 = line + i1;
#pragma unroll
        for (int k0 = 0; k0 < NCH; k0 += 4) {
            int c0 = k0 + 2 * h;
            int c1 = c0 + 1;
            int o0 = c0 * HW;
            int o1 = c1 * HW;
            float s0 = w00 * b00[o0] + w01 * b01[o0] + w10 * b10[o0] + w11 * b11[o0];
            float s1 = w00 * b00[o1] + w01 * b01[o1] + w10 * b10[o1] + w11 * b11[o1];
            float L0 = l0[c0 * RES] * (1.f - wt) + l1[c0 * RES] * wt;
            float L1 = l0[c1 * RES] * (1.f - wt) + l1[c1 * RES] * wt;
            v2f a;
            a.x = s0 * L0;
            a.y = s1 * L1;
            acc = __builtin_amdgcn_wmma_f32_16x16x4_f32(
                false, a, false, ones, (short)0, acc, false, false);
        }
    }
    return acc;
}

// ---------------------------------------------------------------------------
// Main kernel: one wave32 handles 16 points (half-lanes split the channels).
// ---------------------------------------------------------------------------
template <bool PACKED>
__global__ __launch_bounds__(256) void vm_decode_kernel(
    const float* __restrict__ pts,
    const float* __restrict__ pl_yx, const float* __restrict__ pl_zx,
    const float* __restrict__ pl_zy,
    const float* __restrict__ ln_z, const float* __restrict__ ln_y,
    const float* __restrict__ ln_x,
    float* __restrict__ out, int npoints) {
    int lane = threadIdx.x & 31;
    int wave = threadIdx.x >> 5;
    int h    = lane >> 4;        // half: which K-pair group this lane supplies
    int m    = lane & 15;        // A-matrix row = point within group
    int gbase = (blockIdx.x * (int)(blockDim.x >> 5) + wave) * 16;
    int p  = gbase + m;
    int pc = min(p, npoints - 1);   // clamp (not predicate): EXEC must stay all-1s for WMMA

    const float* cp = pts + (size_t)pc * 3;
    float x = cp[0], y = cp[1], z = cp[2];

    v8f acc = {0.f, 0.f, 0.f, 0.f, 0.f, 0.f, 0.f, 0.f};
    v2f ones;
    ones.x = 1.0f;
    ones.y = 1.0f;

    acc = vm_term<PACKED>(acc, ones, pl_yx, ln_z, x, y, z, h);   // f1
    acc = vm_term<PACKED>(acc, ones, pl_zx, ln_y, x, z, y, h);   // f2
    acc = vm_term<PACKED>(acc, ones, pl_zy, ln_x, y, z, x, h);   // f3

    // C/D layout: VGPR r holds row r (lanes 0-15) / row 8+r (lanes 16-31).
    // Lane with column index == r in each half is the designated writer.
#pragma unroll
    for (int r = 0; r < 8; ++r) {
        int pid = gbase + 8 * h + r;
        if (m == r && pid < npoints) out[pid] = acc[r];
    }
}

extern "C" void kernel_launch(void* const* d_in, const int* in_sizes, int n_in,
                              void* d_out, int out_size, void* d_ws, size_t ws_size,
                              hipStream_t stream) {
    const float* in_tensor = (const float*)d_in[0];
    const float* plane_yx  = (const float*)d_in[1];
    const float* plane_zx  = (const float*)d_in[2];
    const float* plane_zy  = (const float*)d_in[3];
    const float* line_z    = (const float*)d_in[4];
    const float* line_y    = (const float*)d_in[5];
    const float* line_x    = (const float*)d_in[6];
    float* out = (float*)d_out;

    int npoints = in_sizes[0] / 3;

    const size_t planeElems = (size_t)NCH * HW;      // 2,160,000
    const size_t lineElems  = (size_t)NCH * RES;     // 7,200
    const size_t needed = (3 * planeElems + 3 * lineElems) * sizeof(float);

    const int points_per_block = (256 / 32) * 16;    // 8 waves * 16 points
    int grid = (npoints + points_per_block - 1) / points_per_block;

    if (ws_size >= needed) {
        float* w0 = (float*)d_ws;
        float* w1 = w0 + planeElems;
        float* w2 = w1 + planeElems;
        float* wl0 = w2 + planeElems;
        float* wl1 = wl0 + lineElems;
        float* wl2 = wl1 + lineElems;

        {
            int n = (int)planeElems, blk = 256, g = (n + blk - 1) / blk;
            vm_repack_plane_kernel<<<g, blk, 0, stream>>>(plane_yx, w0, n);
            vm_repack_plane_kernel<<<g, blk, 0, stream>>>(plane_zx, w1, n);
            vm_repack_plane_kernel<<<g, blk, 0, stream>>>(plane_zy, w2, n);
        }
        {
            int n = (int)lineElems, blk = 256, g = (n + blk - 1) / blk;
            vm_repack_line_kernel<<<g, blk, 0, stream>>>(line_z, wl0, n);
            vm_repack_line_kernel<<<g, blk, 0, stream>>>(line_y, wl1, n);
            vm_repack_line_kernel<<<g, blk, 0, stream>>>(line_x, wl2, n);
        }
        vm_decode_kernel<true><<<grid, 256, 0, stream>>>(in_tensor, w0, w1, w2,
                                                         wl0, wl1, wl2, out, npoints);
    } else {
        vm_decode_kernel<false><<<grid, 256, 0, stream>>>(in_tensor, plane_yx, plane_zx,
                                                          plane_zy, line_z, line_y, line_x,
                                                          out, npoints);
    }
}